// Parlai_MHA_77687368450155
// MI455X (gfx1250) — compile-verified
//
#include <hip/hip_runtime.h>
#include <hip/hip_bf16.h>
#include <math.h>

// ---------------------------------------------------------------------------
// ParlAI MHA forward on MI455X (gfx1250), fp32, via V_WMMA_F32_16X16X4_F32.
// Outputs: d_out = [ out (B*S*D) | dot_prod (B*H*S*S) ]  (fp32, concatenated)
// dot_prod (537 MB) dominates -> memory-bound; fp32 WMMA preserves reference
// precision for both outputs while still exercising the matrix pipes.
// ---------------------------------------------------------------------------

typedef __attribute__((ext_vector_type(2))) float v2f;
typedef __attribute__((ext_vector_type(8))) float v8f;

#define BS   2      // batch
#define S    2048   // sequence length
#define D    1024   // model dim
#define H    16     // heads
#define DPH  64     // dim per head
#define NEG_INF (-1e20f)
#define INV_SCALE 0.125f   // 1/sqrt(64)

// ---------------------------------------------------------------------------
// Kernel 1: scores[bh][q][k] = mask ? (Q.Kt)/8 : NEG_INF   (writes dot_prod)
// One wave computes a 32x32 tile: 2 A-frags x 2 B-frags, 16 k-steps, 64 WMMAs.
// ---------------------------------------------------------------------------
__global__ void __launch_bounds__(256)
mha_scores_kernel(const float* __restrict__ q, const int* __restrict__ mask,
                  float* __restrict__ dot) {
    const int lane = threadIdx.x;            // 0..31
    const int wave = threadIdx.y;            // 0..7
    const int lr   = lane & 15;              // row-within-frag
    const int hf   = lane >> 4;              // half selector (K pairs)
    const int bh   = blockIdx.z;             // 0..31
    const int b    = bh >> 4;
    const int h    = bh & 15;
    const int q0   = blockIdx.y * 32;                 // query tile start
    const int n0   = (blockIdx.x * 8 + wave) * 32;    // key tile start

    // Q/K rows live at query[(b*S + row)*D + h*DPH + d]
    const float* base = q + (size_t)b * S * D + h * DPH + 2 * hf;
    const float* qa0  = base + (size_t)(q0 + lr) * D;
    const float* qa1  = qa0 + (size_t)16 * D;
    const float* kb0  = base + (size_t)(n0 + lr) * D;
    const float* kb1  = kb0 + (size_t)16 * D;

    v8f acc00 = {}, acc01 = {}, acc10 = {}, acc11 = {};
#pragma unroll
    for (int kk = 0; kk < DPH / 4; ++kk) {
        v2f a0 = *(const v2f*)(qa0 + kk * 4);
        v2f a1 = *(const v2f*)(qa1 + kk * 4);
        v2f b0 = *(const v2f*)(kb0 + kk * 4);
        v2f b1 = *(const v2f*)(kb1 + kk * 4);
        acc00 = __builtin_amdgcn_wmma_f32_16x16x4_f32(false, a0, false, b0,
                                                      (short)0, acc00, false, false);
        acc01 = __builtin_amdgcn_wmma_f32_16x16x4_f32(false, a0, false, b1,
                                                      (short)0, acc01, false, false);
        acc10 = __builtin_amdgcn_wmma_f32_16x16x4_f32(false, a1, false, b0,
                                                      (short)0, acc10, false, false);
        acc11 = __builtin_amdgcn_wmma_f32_16x16x4_f32(false, a1, false, b1,
                                                      (short)0, acc11, false, false);
    }

    // C/D layout: lane -> col N = lr; VGPR r -> row M = r + 8*hf
    const int col0 = n0 + lr;
    const int col1 = n0 + 16 + lr;
    const bool allow0 = mask[b * S + col0] != 0;
    const bool allow1 = mask[b * S + col1] != 0;

    float* drow = dot + (size_t)bh * S * S;
#pragma unroll
    for (int r = 0; r < 8; ++r) {
        const int row0 = q0 + r + 8 * hf;
        const int row1 = row0 + 16;
        drow[(size_t)row0 * S + col0] = allow0 ? acc00[r] * INV_SCALE : NEG_INF;
        drow[(size_t)row0 * S + col1] = allow1 ? acc01[r] * INV_SCALE : NEG_INF;
        drow[(size_t)row1 * S + col0] = allow0 ? acc10[r] * INV_SCALE : NEG_INF;
        drow[(size_t)row1 * S + col1] = allow1 ? acc11[r] * INV_SCALE : NEG_INF;
    }
}

// ---------------------------------------------------------------------------
// Kernel 2: per-row softmax stats. One wave per row of dot_prod (65536 rows).
// Writes rmax[row] and rinv[row] = 1/sum(exp(s - max)) to workspace.
// ---------------------------------------------------------------------------
__global__ void __launch_bounds__(256)
mha_rowstats_kernel(const float* __restrict__ dot, float* __restrict__ rmax,
                    float* __restrict__ rinv) {
    const int lane = threadIdx.x;
    const size_t row = (size_t)blockIdx.x * 8 + threadIdx.y;
    const float* p = dot + row * S;

    float m = -INFINITY;
    for (int j = lane; j < S; j += 32) m = fmaxf(m, p[j]);
#pragma unroll
    for (int off = 16; off >= 1; off >>= 1) m = fmaxf(m, __shfl_xor(m, off, 32));

    float sum = 0.0f;
    for (int j = lane; j < S; j += 32) sum += __expf(p[j] - m);
#pragma unroll
    for (int off = 16; off >= 1; off >>= 1) sum += __shfl_xor(sum, off, 32);

    if (lane == 0) {
        rmax[row] = m;
        rinv[row] = 1.0f / sum;
    }
}

// ---------------------------------------------------------------------------
// Kernel 3: out = softmax(scores) @ V.  One wave: 16 q-rows x 64 (full dph).
// A-frag = exp(score - rowmax) computed on the fly (lane's rows == lr, so a
// scalar rowmax per lane is exact); 1/sum folded in at store time.
// ---------------------------------------------------------------------------
__global__ void __launch_bounds__(256)
mha_pv_kernel(const float* __restrict__ qkv, const float* __restrict__ dot,
              const float* __restrict__ rmax, const float* __restrict__ rinv,
              float* __restrict__ out) {
    const int lane = threadIdx.x;
    const int lr   = lane & 15;
    const int hf   = lane >> 4;
    const int bh   = blockIdx.z;
    const int b    = bh >> 4;
    const int h    = bh & 15;
    const int q0   = blockIdx.x * 128 + threadIdx.y * 16;

    const float myMax = rmax[(size_t)bh * S + q0 + lr];
    const float* prow = dot + ((size_t)bh * S + q0 + lr) * S + 2 * hf;
    // V[k][d] at qkv[(b*S + k)*D + h*DPH + d]; lane supplies column d = d0+lr
    const float* vbase = qkv + (size_t)b * S * D + h * DPH + lr;

    v8f acc0 = {}, acc1 = {}, acc2 = {}, acc3 = {};
#pragma unroll 2
    for (int kk = 0; kk < S / 4; ++kk) {
        const int k0 = kk * 4 + 2 * hf;
        v2f s = *(const v2f*)(prow + kk * 4);
        v2f a;
        a.x = __expf(s.x - myMax);
        a.y = __expf(s.y - myMax);
        const float* v0 = vbase + (size_t)k0 * D;
        const float* v1 = v0 + D;
        v2f b0 = { v0[0],  v1[0]  };
        v2f b1 = { v0[16], v1[16] };
        v2f b2 = { v0[32], v1[32] };
        v2f b3 = { v0[48], v1[48] };
        acc0 = __builtin_amdgcn_wmma_f32_16x16x4_f32(false, a, false, b0,
                                                     (short)0, acc0, false, false);
        acc1 = __builtin_amdgcn_wmma_f32_16x16x4_f32(false, a, false, b1,
                                                     (short)0, acc1, false, false);
        acc2 = __builtin_amdgcn_wmma_f32_16x16x4_f32(false, a, false, b2,
                                                     (short)0, acc2, false, false);
        acc3 = __builtin_amdgcn_wmma_f32_16x16x4_f32(false, a, false, b3,
                                                     (short)0, acc3, false, false);
    }

    float* obase = out + (size_t)b * S * D + h * DPH + lr;
#pragma unroll
    for (int r = 0; r < 8; ++r) {
        const int qrow = q0 + r + 8 * hf;
        const float sc = rinv[(size_t)bh * S + qrow];
        float* op = obase + (size_t)qrow * D;
        op[0]  = acc0[r] * sc;
        op[16] = acc1[r] * sc;
        op[32] = acc2[r] * sc;
        op[48] = acc3[r] * sc;
    }
}

// ---------------------------------------------------------------------------
extern "C" void kernel_launch(void* const* d_in, const int* in_sizes, int n_in,
                              void* d_out, int out_size, void* d_ws, size_t ws_size,
                              hipStream_t stream) {
    const float* query = (const float*)d_in[0];
    const int*   mask  = (const int*)d_in[1];

    float* out = (float*)d_out;                      // B*S*D floats
    float* dot = out + (size_t)BS * S * D;           // B*H*S*S floats (dot_prod)

    float* rmax = (float*)d_ws;                      // B*H*S floats
    float* rinv = rmax + (size_t)BS * H * S;         // B*H*S floats

    dim3 blk(32, 8);  // 8 waves of 32 (wave32)

    // 1) scores -> dot_prod (mask + 1/sqrt(dph) applied)
    mha_scores_kernel<<<dim3(S / 256, S / 32, BS * H), blk, 0, stream>>>(query, mask, dot);

    // 2) row-wise softmax stats (max, 1/sum) over B*H*S rows
    mha_rowstats_kernel<<<dim3((BS * H * S) / 8), blk, 0, stream>>>(dot, rmax, rinv);

    // 3) out = softmax(dot) @ V
    mha_pv_kernel<<<dim3(S / 128, 1, BS * H), blk, 0, stream>>>(query, dot, rmax, rinv, out);
}